// E2R_44641890075192
// MI455X (gfx1250) — compile-verified
//
#include <hip/hip_runtime.h>
#include <hip/hip_bf16.h>

typedef float v2f __attribute__((ext_vector_type(2)));
typedef float v4f __attribute__((ext_vector_type(4)));
typedef float v8f __attribute__((ext_vector_type(8)));

#define NN 50000
#define EE 800000
#define EHD 256
#define RHD 128
#define RRR 500

// ---- order-preserving float <-> uint key (for atomic max on floats) ----
__device__ __forceinline__ unsigned f2key(float f) {
  unsigned u = __float_as_uint(f);
  return (u & 0x80000000u) ? ~u : (u | 0x80000000u);
}
__device__ __forceinline__ float key2f(unsigned k) {
  return (k & 0x80000000u) ? __uint_as_float(k & 0x7fffffffu)
                           : __uint_as_float(~k);
}
__device__ __forceinline__ float dot4(v4f a, v4f b) {
  return a.x * b.x + a.y * b.y + a.z * b.z + a.w * b.w;
}

// ============ Kernel A: XH = X @ Wh^T, XT = X @ Wt^T via f32 WMMA ============
// One WG = 16 M-rows. Wave w owns N-columns [16w,16w+16) and computes both
// outputs, reusing the A fragment across the two WMMAs per K-step.
__global__ void k_gemm(const float* __restrict__ X,
                       const float* __restrict__ Wh,
                       const float* __restrict__ Wt,
                       float* __restrict__ XH, float* __restrict__ XT) {
  const int lane = threadIdx.x & 31;
  const int wave = threadIdx.x >> 5;
  const int m0 = blockIdx.x * 16;
  const int n0 = wave * 16;
  const int r  = lane & 15;           // row within 16-tile
  const int kk = (lane >> 4) << 1;    // K sub-offset 0 or 2 (ISA 16x4 layout)

  const float* arow = X  + (size_t)(m0 + r) * EHD + kk;
  const float* brh  = Wh + (size_t)(n0 + r) * EHD + kk;
  const float* brt  = Wt + (size_t)(n0 + r) * EHD + kk;

  v8f ch = {0.f, 0.f, 0.f, 0.f, 0.f, 0.f, 0.f, 0.f};
  v8f ct = {0.f, 0.f, 0.f, 0.f, 0.f, 0.f, 0.f, 0.f};

  for (int k = 0; k < EHD; k += 4) {
    v2f a  = *(const v2f*)(arow + k);
    v2f b0 = *(const v2f*)(brh + k);
    v2f b1 = *(const v2f*)(brt + k);
    ch = __builtin_amdgcn_wmma_f32_16x16x4_f32(false, a, false, b0, (short)0,
                                               ch, false, false);
    ct = __builtin_amdgcn_wmma_f32_16x16x4_f32(false, a, false, b1, (short)0,
                                               ct, false, false);
  }

  // C/D layout: VGPR j, lanes 0-15 -> M=j, lanes 16-31 -> M=j+8; N = lane&15
  const int n  = lane & 15;
  const int mb = (lane >> 4) << 3;
  for (int j = 0; j < 8; ++j) {
    XH[(size_t)(m0 + mb + j) * RHD + n0 + n] = ch[j];
    XT[(size_t)(m0 + mb + j) * RHD + n0 + n] = ct[j];
  }
}

// ============ Kernel B: per-node scores, packed for edge endpoints ==========
// sA[n] = {s_hh0, s_hh1, s_th0, s_th1}  (used when node is src)
// sB[n] = {s_tt0, s_tt1, s_ht0, s_ht1}  (used when node is dst)
__global__ void k_scores(const float* __restrict__ XH,
                         const float* __restrict__ XT,
                         const float* __restrict__ Ah,
                         const float* __restrict__ At,
                         float* __restrict__ sA, float* __restrict__ sB) {
  const int lane = threadIdx.x & 31;
  const int wave = threadIdx.x >> 5;
  const int node = blockIdx.x * 8 + wave;
  const int c = lane * 4;

  v4f xh  = *(const v4f*)(XH + (size_t)node * RHD + c);
  v4f xt  = *(const v4f*)(XT + (size_t)node * RHD + c);
  v4f ah0 = *(const v4f*)(Ah + c);
  v4f ah1 = *(const v4f*)(Ah + RHD + c);
  v4f at0 = *(const v4f*)(At + c);
  v4f at1 = *(const v4f*)(At + RHD + c);

  float v[8];
  v[0] = dot4(xh, ah0);  // s_hh h0
  v[1] = dot4(xh, ah1);  // s_hh h1
  v[2] = dot4(xt, at0);  // s_tt h0
  v[3] = dot4(xt, at1);  // s_tt h1
  v[4] = dot4(xt, ah0);  // s_th h0
  v[5] = dot4(xt, ah1);  // s_th h1
  v[6] = dot4(xh, at0);  // s_ht h0
  v[7] = dot4(xh, at1);  // s_ht h1

  for (int m = 16; m >= 1; m >>= 1)
    for (int j = 0; j < 8; ++j) v[j] += __shfl_xor(v[j], m, 32);

  if (lane == 0) {
    v4f a = {v[0], v[1], v[4], v[5]};
    v4f b = {v[2], v[3], v[6], v[7]};
    *(v4f*)(sA + (size_t)node * 4) = a;
    *(v4f*)(sB + (size_t)node * 4) = b;
  }
}

// ============ Kernel I: init reduction buffers ==============================
__global__ void k_init(unsigned* __restrict__ gkey, float* __restrict__ gsum,
                       float* __restrict__ gacc) {
  int i = blockIdx.x * blockDim.x + threadIdx.x;
  if (i < RRR * 4) { gkey[i] = 0u; gsum[i] = 0.f; }
  if (i < RRR * RHD) gacc[i] = 0.f;
}

// ============ Kernel C: edge logits + segment max ===========================
__global__ void k_edge_logits(const int* __restrict__ EI,
                              const int* __restrict__ rel,
                              const float* __restrict__ sA,
                              const float* __restrict__ sB,
                              float* __restrict__ eV,
                              unsigned* __restrict__ gkey) {
  int e = blockIdx.x * blockDim.x + threadIdx.x;
  if (e >= EE) return;
  int src = EI[e], dst = EI[EE + e], r = rel[e];
  v4f sa = *(const v4f*)(sA + (size_t)src * 4);
  v4f sb = *(const v4f*)(sB + (size_t)dst * 4);
  v4f ev;
  float t;
  t = sa.x + sb.x; ev.x = t > 0.f ? t : 0.01f * t;  // e1 head0
  t = sa.y + sb.y; ev.y = t > 0.f ? t : 0.01f * t;  // e1 head1
  t = sa.z + sb.z; ev.z = t > 0.f ? t : 0.01f * t;  // e2 head0
  t = sa.w + sb.w; ev.w = t > 0.f ? t : 0.01f * t;  // e2 head1
  *(v4f*)(eV + (size_t)e * 4) = ev;
  unsigned* g = gkey + r * 4;
  atomicMax(&g[0], f2key(ev.x));
  atomicMax(&g[1], f2key(ev.y));
  atomicMax(&g[2], f2key(ev.z));
  atomicMax(&g[3], f2key(ev.w));
}

// ============ Kernel D: exp(e - max) + segment sum ==========================
__global__ void k_edge_exp(const int* __restrict__ rel, float* __restrict__ eV,
                           const unsigned* __restrict__ gkey,
                           float* __restrict__ gsum) {
  int e = blockIdx.x * blockDim.x + threadIdx.x;
  if (e >= EE) return;
  int r = rel[e];
  v4f ev = *(const v4f*)(eV + (size_t)e * 4);
  float m0 = key2f(gkey[r * 4 + 0]); if (!__builtin_isfinite(m0)) m0 = 0.f;
  float m1 = key2f(gkey[r * 4 + 1]); if (!__builtin_isfinite(m1)) m1 = 0.f;
  float m2 = key2f(gkey[r * 4 + 2]); if (!__builtin_isfinite(m2)) m2 = 0.f;
  float m3 = key2f(gkey[r * 4 + 3]); if (!__builtin_isfinite(m3)) m3 = 0.f;
  v4f ex;
  ex.x = expf(ev.x - m0);
  ex.y = expf(ev.y - m1);
  ex.z = expf(ev.z - m2);
  ex.w = expf(ev.w - m3);
  *(v4f*)(eV + (size_t)e * 4) = ex;  // overwrite in place (recomputed per call)
  atomicAdd(&gsum[r * 4 + 0], ex.x);
  atomicAdd(&gsum[r * 4 + 1], ex.y);
  atomicAdd(&gsum[r * 4 + 2], ex.z);
  atomicAdd(&gsum[r * 4 + 3], ex.w);
}

// ============ Kernel E: weighted segment-sum with LDS-resident accumulator ==
// The full [500][128] f32 accumulator (256 KB) lives in LDS per workgroup
// (fits in the 320 KB WGP LDS). One wave processes one edge at a time:
// lane l handles feature columns [4l, 4l+4).
__global__ void k_aggregate(const int* __restrict__ EI,
                            const int* __restrict__ rel,
                            const float* __restrict__ eV,
                            const float* __restrict__ gsum,
                            const float* __restrict__ XH,
                            const float* __restrict__ XT,
                            float* __restrict__ gacc) {
  extern __shared__ float lacc[];
  const int ACC = RRR * RHD;  // 64000 floats
  for (int i = threadIdx.x; i < ACC; i += blockDim.x) lacc[i] = 0.f;
  __syncthreads();

  const int lane = threadIdx.x & 31;
  const int wave = threadIdx.x >> 5;
  const int gw = blockIdx.x * (blockDim.x >> 5) + wave;
  const int nw = gridDim.x * (blockDim.x >> 5);
  const int c = lane * 4;

  for (int e = gw; e < EE; e += nw) {
    int src = EI[e], dst = EI[EE + e], r = rel[e];
    v4f ex = *(const v4f*)(eV + (size_t)e * 4);
    const float* s = gsum + r * 4;
    float wA = ex.x / (s[0] + 1e-16f) + ex.y / (s[1] + 1e-16f);
    float wB = ex.z / (s[2] + 1e-16f) + ex.w / (s[3] + 1e-16f);
    v4f xh = *(const v4f*)(XH + (size_t)src * RHD + c);
    v4f xt = *(const v4f*)(XT + (size_t)dst * RHD + c);
    float* d = lacc + r * RHD + c;
    atomicAdd(&d[0], wA * xh.x + wB * xt.x);
    atomicAdd(&d[1], wA * xh.y + wB * xt.y);
    atomicAdd(&d[2], wA * xh.z + wB * xt.z);
    atomicAdd(&d[3], wA * xh.w + wB * xt.w);
  }
  __syncthreads();

  for (int i = threadIdx.x; i < ACC; i += blockDim.x) {
    float v = lacc[i];
    if (v != 0.f) atomicAdd(&gacc[i], v);
  }
}

// ============ Kernel F: scale and write output ==============================
__global__ void k_final(const float* __restrict__ gacc, float* __restrict__ out) {
  int i = blockIdx.x * blockDim.x + threadIdx.x;
  if (i < RRR * RHD) out[i] = gacc[i] * 0.25f;  // / num_heads / 2
}

extern "C" void kernel_launch(void* const* d_in, const int* in_sizes, int n_in,
                              void* d_out, int out_size, void* d_ws,
                              size_t ws_size, hipStream_t stream) {
  const float* x_e = (const float*)d_in[0];
  const int* edge_index = (const int*)d_in[1];
  const int* rel = (const int*)d_in[2];
  const float* w_h = (const float*)d_in[3];
  const float* w_t = (const float*)d_in[4];
  const float* a_h = (const float*)d_in[5];
  const float* a_t = (const float*)d_in[6];
  float* out = (float*)d_out;

  // workspace carve-up (~66 MB)
  float* XH = (float*)d_ws;                       // N*128
  float* XT = XH + (size_t)NN * RHD;              // N*128
  float* sA = XT + (size_t)NN * RHD;              // N*4
  float* sB = sA + (size_t)NN * 4;                // N*4
  float* eV = sB + (size_t)NN * 4;                // E*4
  unsigned* gkey = (unsigned*)(eV + (size_t)EE * 4);  // R*4
  float* gsum = (float*)(gkey + RRR * 4);         // R*4
  float* gacc = gsum + RRR * 4;                   // R*128

  k_gemm<<<NN / 16, 256, 0, stream>>>(x_e, w_h, w_t, XH, XT);
  k_scores<<<NN / 8, 256, 0, stream>>>(XH, XT, a_h, a_t, sA, sB);
  k_init<<<(RRR * RHD + 255) / 256, 256, 0, stream>>>(gkey, gsum, gacc);
  k_edge_logits<<<(EE + 255) / 256, 256, 0, stream>>>(edge_index, rel, sA, sB,
                                                      eV, gkey);
  k_edge_exp<<<(EE + 255) / 256, 256, 0, stream>>>(rel, eV, gkey, gsum);
  k_aggregate<<<256, 256, RRR * RHD * sizeof(float), stream>>>(
      edge_index, rel, eV, gsum, XH, XT, gacc);
  k_final<<<(RRR * RHD + 255) / 256, 256, 0, stream>>>(gacc, out);
}